// CrossAttention_89232240542671
// MI455X (gfx1250) — compile-verified
//
#include <hip/hip_runtime.h>
#include <hip/hip_bf16.h>
#include <math.h>

// ---------------------------------------------------------------------------
// Cross-covariance attention, fp32, CDNA5 (gfx1250) WMMA f32 16x16x4 path.
//
// Algebraic fold:  y[b] = Weff[b] @ x_k[b] + beff[b],
//   Weff[b] = Wp @ blockdiag_h(attn[b,h]) @ Wv   (192x192, per batch)
// so v / out tensors are never materialized (saves ~200MB HBM traffic).
// ---------------------------------------------------------------------------

typedef float v2f __attribute__((ext_vector_type(2)));
typedef float v8f __attribute__((ext_vector_type(8)));

#define HW    16384
#define CC    192
#define NHD   4
#define HD    48
#define PIX   64      // pixels per workgroup tile
#define LSTR  68      // LDS row stride (floats): 16B aligned, conflict-free
#define TSTR  196     // phase2 T stride (192 would alias LDS banks)

__device__ __forceinline__ v8f vzero8() {
  v8f z = {0.f, 0.f, 0.f, 0.f, 0.f, 0.f, 0.f, 0.f};
  return z;
}

// D(16x16,f32) += A(16x4,f32) x B(4x16,f32)
// A layout: lanes 0-15 M=lane {K0,K1}, lanes 16-31 M=lane-16 {K2,K3}
// B layout: lanes 0-15 N=lane rows {K0,K1}, lanes 16-31 N=lane-16 rows {K2,K3}
__device__ __forceinline__ v8f wmma_f32(v2f a, v2f b, v8f c) {
  return __builtin_amdgcn_wmma_f32_16x16x4_f32(false, a, false, b,
                                               (short)0, c, false, false);
}

// ===========================================================================
// Phase 1: q = Wq x_q + bq, k = Wk x_k + bk (WMMA GEMM), per-pixel/head L2
// norms folded into k, then Gram S[b,h] += q_h * k_h^T over the pixel tile.
// grid = B * (HW/PIX) = 2048, block = 256 (8 waves)
// ===========================================================================
__global__ void __launch_bounds__(256, 1)
phase1_qk_gram(const float* __restrict__ x_q, const float* __restrict__ x_k,
               const float* __restrict__ Wq, const float* __restrict__ bq,
               const float* __restrict__ Wkv, const float* __restrict__ bkv,
               float* __restrict__ S) {
  extern __shared__ float sm[];
  float* xq  = sm;                 // 192*68
  float* xk  = xq + CC * LSTR;     // 192*68
  float* qb  = xk + CC * LSTR;     // 192*68
  float* kb  = qb + CC * LSTR;     // 192*68
  float* inv = kb + CC * LSTR;     // 512: [mat][h][d]

  const int tid   = threadIdx.x;
  const int lane  = tid & 31;
  const int wave  = tid >> 5;
  const int lidx  = lane & 15;     // M (A) / N (B,C) index
  const int lhalf = lane >> 4;     // 0: K0/K1,M+0  1: K2/K3,M+8
  const int koff  = lhalf << 1;

  const int b  = blockIdx.x >> 8;            // 256 tiles per batch
  const int d0 = (blockIdx.x & 255) * PIX;

  const float* gq = x_q + ((size_t)b * CC) * HW + d0;
  const float* gk = x_k + ((size_t)b * CC) * HW + d0;

  // ---- stage x tiles (192 x 64) into LDS, float4 ----
  for (int i = tid; i < CC * (PIX / 4); i += 256) {
    int c = i >> 4, f = (i & 15) << 2;
    *(float4*)(xq + c * LSTR + f) = *(const float4*)(gq + (size_t)c * HW + f);
    *(float4*)(xk + c * LSTR + f) = *(const float4*)(gk + (size_t)c * HW + f);
  }
  __syncthreads();

  // ---- q / k GEMM: 24 row-strips (12 per matrix), 4 col-tiles each ----
  for (int strip = wave; strip < 24; strip += 8) {
    const int mat = strip / 12;            // 0 = q, 1 = k
    const int rt  = strip % 12;
    const float* W    = mat ? Wkv : Wq;    // k uses rows 0..191 of Wkv
    const float* bias = mat ? bkv : bq;
    const float* xl   = mat ? xk : xq;
    float*       ol   = mat ? kb : qb;

    v8f acc[4];
    #pragma unroll
    for (int ct = 0; ct < 4; ++ct) acc[ct] = vzero8();

    const float* wrow = W + (size_t)(rt * 16 + lidx) * CC;
    for (int k0 = 0; k0 < CC; k0 += 4) {
      v2f a;
      a.x = wrow[k0 + koff];
      a.y = wrow[k0 + koff + 1];
      #pragma unroll
      for (int ct = 0; ct < 4; ++ct) {
        v2f bm;
        bm.x = xl[(k0 + koff) * LSTR + ct * 16 + lidx];
        bm.y = xl[(k0 + koff + 1) * LSTR + ct * 16 + lidx];
        acc[ct] = wmma_f32(a, bm, acc[ct]);
      }
    }
    #pragma unroll
    for (int ct = 0; ct < 4; ++ct) {
      #pragma unroll
      for (int i = 0; i < 8; ++i) {
        int row = rt * 16 + i + (lhalf << 3);
        ol[row * LSTR + ct * 16 + lidx] = acc[ct][i] + bias[row];
      }
    }
  }
  __syncthreads();

  // ---- per-pixel, per-head L2 norms of q and k (over 48 channels) ----
  for (int item = tid; item < 2 * NHD * PIX; item += 256) {
    int mat = item >> 8, rest = item & 255, h = rest >> 6, d = rest & 63;
    const float* src = mat ? kb : qb;
    float s = 0.f;
    #pragma unroll 4
    for (int j = 0; j < HD; ++j) {
      float v = src[(h * HD + j) * LSTR + d];
      s += v * v;
    }
    inv[item] = 1.0f / fmaxf(sqrtf(s), 1e-12f);   // torch F.normalize semantics
  }
  __syncthreads();

  // ---- fold 1/(|q||k|) into k columns (per head of the row's channel) ----
  for (int e = tid; e < CC * PIX; e += 256) {
    int c = e >> 6, d = e & 63, h = c / HD;
    kb[c * LSTR + d] *= inv[h * 64 + d] * inv[256 + h * 64 + d];
  }
  __syncthreads();

  // ---- Gram: S[b,h](48x48) += q_h(48xP) * k_h(48xP)^T, WMMA over pixels ----
  for (int t = wave; t < NHD * 9; t += 8) {
    int h = t / 9, rr = t % 9, rt = rr / 3, ct = rr % 3;
    v8f acc = vzero8();
    const float* arow = qb + (h * HD + rt * 16 + lidx) * LSTR;
    const float* brow = kb + (h * HD + ct * 16 + lidx) * LSTR;
    for (int p0 = 0; p0 < PIX; p0 += 4) {
      v2f a, bm;
      a.x  = arow[p0 + koff];
      a.y  = arow[p0 + koff + 1];
      bm.x = brow[p0 + koff];       // B[p][n] = k[cbase+n][p]
      bm.y = brow[p0 + koff + 1];
      acc = wmma_f32(a, bm, acc);
    }
    float* Sh = S + ((size_t)(b * NHD + h) * HD) * HD;
    #pragma unroll
    for (int i = 0; i < 8; ++i) {
      int c  = rt * 16 + i + (lhalf << 3);
      int kk = ct * 16 + lidx;
      unsafeAtomicAdd(&Sh[c * HD + kk], acc[i]);   // global_atomic_add_f32
    }
  }
}

// ===========================================================================
// Phase 2 (tiny): attn = softmax(scale_h * S); T = blockdiag(attn) @ Wv;
// Weff[b] = Wp @ T; beff[b] = Wp @ (attn @ bv) + bp.
// grid = (B, 4): blockIdx.y selects 48 output rows of Weff.
// ===========================================================================
__global__ void __launch_bounds__(256, 1)
phase2_weff(const float* __restrict__ S, const float* __restrict__ Wkv,
            const float* __restrict__ bkv, const float* __restrict__ Wp,
            const float* __restrict__ bp, const float* __restrict__ temp,
            float* __restrict__ Weff, float* __restrict__ beff) {
  extern __shared__ float sm[];
  float* attn = sm;                 // 192*48  rows C = h*48+c
  float* T    = attn + CC * HD;     // 192*TSTR
  float* u    = T + CC * TSTR;      // 192
  const int tid   = threadIdx.x;
  const int b     = blockIdx.x;
  const int chunk = blockIdx.y;     // rows o in [chunk*48, chunk*48+48)

  if (tid < CC) {
    int h = tid / HD;
    float sc = 0.1f / (1.0f + __expf(-temp[h]));   // 0.1*sigmoid(t)
    const float* srow = S + (((size_t)b * NHD + h) * HD + (tid % HD)) * HD;
    float m = -1e30f;
    for (int k = 0; k < HD; ++k) m = fmaxf(m, srow[k] * sc);
    float sum = 0.f;
    for (int k = 0; k < HD; ++k) {
      float e = __expf(srow[k] * sc - m);
      attn[tid * HD + k] = e;
      sum += e;
    }
    float rinv = 1.0f / sum;
    float uu = 0.f;
    for (int k = 0; k < HD; ++k) {
      float a = attn[tid * HD + k] * rinv;
      attn[tid * HD + k] = a;
      uu += a * bkv[CC + h * HD + k];              // bv
    }
    u[tid] = uu;
  }
  __syncthreads();

  // T[C][i] = sum_k attn[C][k] * Wv[(C/48)*48 + k][i]
  for (int e = tid; e < CC * CC; e += 256) {
    int C = e / CC, i = e % CC;
    const float* arow = attn + C * HD;
    const float* wv = Wkv + ((size_t)(CC + (C / HD) * HD)) * CC + i;
    float acc = 0.f;
    for (int k = 0; k < HD; ++k) acc += arow[k] * wv[(size_t)k * CC];
    T[C * TSTR + i] = acc;
  }
  __syncthreads();

  // Weff[b][o][i] = sum_C Wp[o][C] * T[C][i]
  for (int e = tid; e < HD * CC; e += 256) {
    int o = chunk * HD + e / CC, i = e % CC;
    const float* wr = Wp + (size_t)o * CC;
    float acc = 0.f;
    for (int C = 0; C < CC; ++C) acc += wr[C] * T[C * TSTR + i];
    Weff[((size_t)b * CC + o) * CC + i] = acc;
  }
  if (tid < HD) {
    int o = chunk * HD + tid;
    const float* wr = Wp + (size_t)o * CC;
    float acc = bp[o];
    for (int C = 0; C < CC; ++C) acc += wr[C] * u[C];
    beff[b * CC + o] = acc;
  }
}

// ===========================================================================
// Phase 3: out[b] = Weff[b] @ x_k[b] + beff[b]   (x_k should be L2-resident)
// grid = B * (HW/PIX) = 2048, block = 256
// ===========================================================================
__global__ void __launch_bounds__(256, 1)
phase3_out(const float* __restrict__ x_k, const float* __restrict__ Weff,
           const float* __restrict__ beff, float* __restrict__ out) {
  extern __shared__ float sm[];
  float* xk = sm;                  // 192*68
  const int tid   = threadIdx.x;
  const int lane  = tid & 31;
  const int wave  = tid >> 5;
  const int lidx  = lane & 15;
  const int lhalf = lane >> 4;
  const int koff  = lhalf << 1;
  const int b  = blockIdx.x >> 8;
  const int d0 = (blockIdx.x & 255) * PIX;

  const float* gk = x_k + ((size_t)b * CC) * HW + d0;
  for (int i = tid; i < CC * (PIX / 4); i += 256) {
    int c = i >> 4, f = (i & 15) << 2;
    *(float4*)(xk + c * LSTR + f) = *(const float4*)(gk + (size_t)c * HW + f);
  }
  __syncthreads();

  const float* W  = Weff + (size_t)b * CC * CC;
  const float* bb = beff + b * CC;

  // 24 half-strips (12 row-tiles x 2 col-pairs) over 8 waves: balanced
  for (int s = wave; s < 24; s += 8) {
    int rt = s >> 1, cp = (s & 1) << 1;    // col tiles cp, cp+1
    v8f acc0 = vzero8(), acc1 = vzero8();
    const float* wrow = W + (size_t)(rt * 16 + lidx) * CC;
    for (int k0 = 0; k0 < CC; k0 += 4) {
      v2f a;
      a.x = wrow[k0 + koff];
      a.y = wrow[k0 + koff + 1];
      v2f b0, b1;
      b0.x = xk[(k0 + koff) * LSTR + cp * 16 + lidx];
      b0.y = xk[(k0 + koff + 1) * LSTR + cp * 16 + lidx];
      b1.x = xk[(k0 + koff) * LSTR + (cp + 1) * 16 + lidx];
      b1.y = xk[(k0 + koff + 1) * LSTR + (cp + 1) * 16 + lidx];
      acc0 = wmma_f32(a, b0, acc0);
      acc1 = wmma_f32(a, b1, acc1);
    }
    #pragma unroll
    for (int i = 0; i < 8; ++i) {
      int row = rt * 16 + i + (lhalf << 3);
      float bias = bb[row];
      size_t base = ((size_t)b * CC + row) * HW + d0;
      out[base + cp * 16 + lidx]       = acc0[i] + bias;
      out[base + (cp + 1) * 16 + lidx] = acc1[i] + bias;
    }
  }
}

// ===========================================================================
extern "C" void kernel_launch(void* const* d_in, const int* in_sizes, int n_in,
                              void* d_out, int out_size, void* d_ws, size_t ws_size,
                              hipStream_t stream) {
  (void)in_sizes; (void)n_in; (void)out_size; (void)ws_size;
  const float* x_q  = (const float*)d_in[0];
  const float* x_k  = (const float*)d_in[1];
  const float* Wq   = (const float*)d_in[2];
  const float* bq   = (const float*)d_in[3];
  const float* Wkv  = (const float*)d_in[4];
  const float* bkv  = (const float*)d_in[5];
  const float* Wp   = (const float*)d_in[6];
  const float* bp   = (const float*)d_in[7];
  const float* temp = (const float*)d_in[8];
  float* out = (float*)d_out;

  const int B = 8;
  float* S    = (float*)d_ws;                   // B*NH*48*48 = 73728 f
  float* Weff = S + (size_t)B * NHD * HD * HD;  // B*192*192  = 294912 f
  float* beff = Weff + (size_t)B * CC * CC;     // B*192

  hipMemsetAsync(S, 0, (size_t)B * NHD * HD * HD * sizeof(float), stream);

  size_t lds1 = (size_t)(4 * CC * LSTR + 512) * sizeof(float);     // ~206 KB
  phase1_qk_gram<<<B * (HW / PIX), 256, lds1, stream>>>(
      x_q, x_k, Wq, bq, Wkv, bkv, S);

  size_t lds2 = (size_t)(CC * HD + CC * TSTR + CC) * sizeof(float); // ~184 KB
  phase2_weff<<<dim3(B, 4), 256, lds2, stream>>>(
      S, Wkv, bkv, Wp, bp, temp, Weff, beff);

  size_t lds3 = (size_t)(CC * LSTR) * sizeof(float);                // ~51 KB
  phase3_out<<<B * (HW / PIX), 256, lds3, stream>>>(x_k, Weff, beff, out);
}